// AttGRUcell_42820823941243
// MI455X (gfx1250) — compile-verified
//
#include <hip/hip_runtime.h>
#include <cstdint>
#include <cstddef>

typedef __attribute__((ext_vector_type(2))) float f32x2;
typedef __attribute__((ext_vector_type(8))) float f32x8;

#define NN       4096
#define CHUNK    64                 // K elements per pipeline stage
#define LSTRIDE  68                 // padded LDS row stride (floats): banks 4n mod 64, conflict-free
#define PIPE     3                  // async ring depth
#define KHALF    2048               // K split across wave pairs
#define NC       (KHALF / CHUNK)    // 32 stages per wave
#define TILE_F   (16 * LSTRIDE)     // 1088 floats per stage buffer
#define EPI_OFF  (8 * PIPE * TILE_F)
#define SMEM_F   (EPI_OFF + 8 * 128)
#define SMEM_BYTES (SMEM_F * 4)     // ~106 KB dynamic LDS

// Async global->LDS, 16B/lane (512B per instruction per wave). GV mode.
__device__ __forceinline__ void async_b128(unsigned lds_addr, const float* g) {
    unsigned long long ga = (unsigned long long)(uintptr_t)g;
    asm volatile("global_load_async_to_lds_b128 %0, %1, off"
                 :: "v"(lds_addr), "v"(ga) : "memory");
}
// Flat pointer low 32 bits == wave-relative LDS address (CDNA5 aperture mapping).
__device__ __forceinline__ unsigned lds_off(const void* p) {
    return (unsigned)(uintptr_t)p;
}

__global__ __launch_bounds__(256)
void attgru_fused_kernel(const float* __restrict__ x,   const float* __restrict__ h,
                         const float* __restrict__ adj,
                         const float* __restrict__ Whr, const float* __restrict__ bhr,
                         const float* __restrict__ Whz, const float* __restrict__ bhz,
                         const float* __restrict__ Whn, const float* __restrict__ bhn,
                         float* __restrict__ out)
{
    extern __shared__ float smem[];

    const int tid  = threadIdx.x;
    const int wave = tid >> 5;          // 0..7
    const int lane = tid & 31;
    const int mid  = wave & 3;          // 0:adj/x 1:Whr/h 2:Whz/h 3:Whn/h
    const int kh   = wave >> 2;         // K half: [kh*2048, kh*2048+2048)
    const int c0   = blockIdx.x * 16;   // output-column tile

    const float* Wm  = (mid == 0) ? adj : (mid == 1) ? Whr : (mid == 2) ? Whz : Whn;
    const float* act = (mid == 0) ? x : h;
    const int    k0b = kh * KHALF;

    float* wt_base = smem + wave * (PIPE * TILE_F);

    // Fragment addressing (f32 16x16x4 layouts):
    //   A 16x4: lane<16 -> (m=lane, k,k+1) ; lane>=16 -> (m=lane-16, k+2,k+3)
    //   B 4x16: lane<16 -> (n=lane, k,k+1) ; lane>=16 -> (n=lane-16, k+2,k+3)
    const int  half  = lane >> 4;
    const int  rsel  = lane & 15;
    const bool valid = rsel < 8;                      // batch rows 8..15 are zero padding
    const int  fbase = rsel * LSTRIDE + 2 * half;
    const float* abase = act + (size_t)(rsel & 7) * NN + k0b + 2 * half;  // L2-resident

    // One async stage: 16 weight rows x 64 floats = 256 x 16B transfers = 8 instructions.
    auto issue = [&](int ch) {
        float* wt = wt_base + (ch % PIPE) * TILE_F;
        const int k0 = k0b + ch * CHUNK;
        asm volatile("s_wait_dscnt 0" ::: "memory");  // recycled ring buffer fully read
        #pragma unroll
        for (int i = 0; i < 8; ++i) {
            int t = i * 32 + lane;
            int row = t >> 4, seg = t & 15;
            async_b128(lds_off(wt + row * LSTRIDE + seg * 4),
                       Wm + ((size_t)(c0 + row) * NN + k0 + seg * 4));
        }
    };

    f32x8 acc = {0.f, 0.f, 0.f, 0.f, 0.f, 0.f, 0.f, 0.f};

    issue(0);
    issue(1);
    for (int ch = 0; ch < NC; ++ch) {
        const float* wt = wt_base + (ch % PIPE) * TILE_F;
        if (ch + 2 < NC) {
            issue(ch + 2);                                    // 24 in flight
            asm volatile("s_wait_asynccnt 16" ::: "memory");  // -> stage ch complete
        } else if (ch + 1 < NC) {
            asm volatile("s_wait_asynccnt 8" ::: "memory");
        } else {
            asm volatile("s_wait_asynccnt 0" ::: "memory");
        }
        #pragma unroll
        for (int kk = 0; kk < CHUNK; kk += 4) {
            f32x2 b  = *(const f32x2*)(wt + fbase + kk);               // ds_load_b64
            f32x2 av = *(const f32x2*)(abase + ch * CHUNK + kk);       // global b64, cache hit
            f32x2 a;
            a.x = valid ? av.x : 0.0f;                                 // zero batch padding,
            a.y = valid ? av.y : 0.0f;                                 // EXEC stays uniform
            acc = __builtin_amdgcn_wmma_f32_16x16x4_f32(
                false, a, false, b, (short)0, acc, false, false);
        }
    }

    // Stash each wave's 8x16 valid tile (VGPR v = row m=v, lanes 0..15 = n).
    float* epi = smem + EPI_OFF;
    if (lane < 16) {
        #pragma unroll
        for (int v = 0; v < 8; ++v) epi[wave * 128 + v * 16 + lane] = acc[v];
    }
    __syncthreads();

    // Fused gate epilogue: sum the two K-halves per matrix, apply GRU math.
    if (tid < 128) {
        const int m = tid >> 4, n = tid & 15, c = c0 + n;
        float agg = epi[0 * 128 + m * 16 + n] + epi[4 * 128 + m * 16 + n];
        float rl  = epi[1 * 128 + m * 16 + n] + epi[5 * 128 + m * 16 + n] + bhr[c];
        float zl  = epi[2 * 128 + m * 16 + n] + epi[6 * 128 + m * 16 + n] + bhz[c];
        float nl  = epi[3 * 128 + m * 16 + n] + epi[7 * 128 + m * 16 + n] + bhn[c];
        float hv  = h[(size_t)m * NN + c];
        float r  = 1.0f / (1.0f + expf(-(agg + rl)));
        float z  = 1.0f / (1.0f + expf(-(agg + zl)));
        float cn = tanhf(agg + r * nl);
        out[(size_t)m * NN + c] = (1.0f - z) * cn + z * hv;
    }
}

extern "C" void kernel_launch(void* const* d_in, const int* in_sizes, int n_in,
                              void* d_out, int out_size, void* d_ws, size_t ws_size,
                              hipStream_t stream) {
    const float* x   = (const float*)d_in[0];
    const float* h   = (const float*)d_in[1];
    const float* adj = (const float*)d_in[2];
    const float* Whr = (const float*)d_in[3];
    const float* bhr = (const float*)d_in[4];
    const float* Whz = (const float*)d_in[5];
    const float* bhz = (const float*)d_in[6];
    const float* Whn = (const float*)d_in[7];
    const float* bhn = (const float*)d_in[8];

    attgru_fused_kernel<<<dim3(NN / 16), dim3(256), SMEM_BYTES, stream>>>(
        x, h, adj, Whr, bhr, Whz, bhz, Whn, bhn, (float*)d_out);
}